// Block_4904852652115
// MI455X (gfx1250) — compile-verified
//
#include <hip/hip_runtime.h>
#include <math.h>
#include <stdint.h>

// ---------------- problem constants (match reference) ----------------
#define NN   50000      // nodes
#define NE   800000     // edges
#define DIM  64
#define EDIM 16
#define HEADS 4
#define HC   256        // HEADS*DIM
#define GDIM 192        // 3*DIM
#define NEG_SLOPE 0.2f

typedef __attribute__((ext_vector_type(2))) float v2f;
typedef __attribute__((ext_vector_type(8))) float v8f;

// ---------------- ordered-float encoding for integer atomicMax -------
__device__ __forceinline__ unsigned f32_ord(float f) {
  unsigned u = __float_as_uint(f);
  return u ^ ((unsigned)(((int)u) >> 31) | 0x80000000u);
}
__device__ __forceinline__ float ord_f32(unsigned v) {
  unsigned u = v ^ ((unsigned)((~(int)v) >> 31) | 0x80000000u);
  return __uint_as_float(u);
}

// ---------------- fp32 WMMA GEMM: C[16m x 16n] tiles -----------------
// One wave (32 threads) per 16x16 output tile. K stepped by 4 using
// V_WMMA_F32_16X16X4_F32.  A: [M,K] row-major.  B: [K,NC] row-major,
// or [NC,K] row-major when TRANSB (i.e. C = A @ B^T).
template<int K, int NC, bool TRANSB, bool BIAS_CELU>
__global__ void gemm16_f32(const float* __restrict__ A,
                           const float* __restrict__ B,
                           const float* __restrict__ bias,
                           float* __restrict__ C) {
  const int lane = threadIdx.x & 31;
  const int ml   = lane & 15;       // row (A) / col (B,C) within tile
  const int half = lane >> 4;       // K sub-pair selector
  const int m0   = blockIdx.x * 16;
  const int n0   = blockIdx.y * 16;

  v8f acc = {};
  const float* __restrict__ arow = A + (size_t)(m0 + ml) * K + half * 2;
  for (int k = 0; k < K; k += 4) {
    v2f a, b;
    a.x = arow[k];
    a.y = arow[k + 1];
    if (TRANSB) {
      const float* bp = B + (size_t)(n0 + ml) * K + k + half * 2;
      b.x = bp[0];
      b.y = bp[1];
    } else {
      const float* bp = B + (size_t)(k + half * 2) * NC + n0 + ml;
      b.x = bp[0];
      b.y = bp[NC];
    }
    acc = __builtin_amdgcn_wmma_f32_16x16x4_f32(
        /*neg_a=*/false, a, /*neg_b=*/false, b,
        /*c_mod=*/(short)0, acc, /*reuse_a=*/false, /*reuse_b=*/false);
  }
#pragma unroll
  for (int v = 0; v < 8; ++v) {
    int m = m0 + v + half * 8;
    int n = n0 + ml;
    float val = acc[v];
    if (BIAS_CELU) {
      val += bias[n];
      val = (val > 0.0f) ? val : (expf(val) - 1.0f);   // celu, alpha=1
    }
    C[(size_t)m * NC + n] = val;
  }
}

// ---------------- fold We with wa[:,1,:] : Wea[k][h] -----------------
__global__ void fold_wea(const float* __restrict__ We,    // [16,256]
                         const float* __restrict__ Wa,    // [H,3,64] flat
                         float* __restrict__ Wea) {       // [16,4]
  int t = threadIdx.x;              // 64 threads: t = k*4 + h
  if (t >= EDIM * HEADS) return;
  int k = t >> 2, h = t & 3;
  float s = 0.0f;
  for (int c = 0; c < DIM; ++c)
    s += We[k * HC + h * DIM + c] * Wa[h * GDIM + DIM + c];
  Wea[k * HEADS + h] = s;
}

// ---------------- per-node attention dots ai/aj ----------------------
__global__ void node_alpha(const float* __restrict__ xp,  // [N,256]
                           const float* __restrict__ Wa,  // [H,3,64]
                           float* __restrict__ ai,        // [N,4]
                           float* __restrict__ aj) {      // [N,4]
  int idx = blockIdx.x * blockDim.x + threadIdx.x;
  if (idx >= NN * HEADS) return;
  int n = idx >> 2, h = idx & 3;
  const float* row = xp + (size_t)n * HC + h * DIM;
  const float* w0 = Wa + h * GDIM;            // wa[h,0,:]
  const float* w2 = Wa + h * GDIM + 2 * DIM;  // wa[h,2,:]
  float si = 0.0f, sj = 0.0f;
  for (int c = 0; c < DIM; ++c) {
    float v = row[c];
    si += v * w0[c];
    sj += v * w2[c];
  }
  ai[idx] = si;
  aj[idx] = sj;
}

// ---------------- per-step init: aggr=0, denom=0, amax=-inf ----------
__global__ void init_step(float* __restrict__ aggr,       // [N,256]
                          float* __restrict__ denom,      // [N,4]
                          unsigned* __restrict__ amax) {  // [N,4]
  int idx = blockIdx.x * blockDim.x + threadIdx.x;
  if (idx < NN * HC) aggr[idx] = 0.0f;
  if (idx < NN * HEADS) {
    denom[idx] = 0.0f;
    amax[idx] = f32_ord(-INFINITY);
  }
}

// ---------------- pass A: raw logits + segment max -------------------
__global__ void edge_alpha(const int* __restrict__ src,
                           const int* __restrict__ dst,
                           const float* __restrict__ ea,   // [E,16]
                           const float* __restrict__ Wea,  // [16,4]
                           const float* __restrict__ ai,
                           const float* __restrict__ aj,
                           float* __restrict__ alphaw,     // [E,4]
                           unsigned* __restrict__ amax) {
  int idx = blockIdx.x * blockDim.x + threadIdx.x;
  if (idx >= NE * HEADS) return;
  int e = idx >> 2, h = idx & 3;
  int s = src[e], d = dst[e];
  float a = ai[d * HEADS + h] + aj[s * HEADS + h];
  const float* er = ea + (size_t)e * EDIM;
#pragma unroll
  for (int k = 0; k < EDIM; ++k) a += er[k] * Wea[k * HEADS + h];
  a = (a > 0.0f) ? a : NEG_SLOPE * a;          // leaky relu
  alphaw[idx] = a;
  atomicMax(&amax[d * HEADS + h], f32_ord(a));
}

// ---------------- pass B: exp + segment sum --------------------------
__global__ void edge_exp(const int* __restrict__ dst,
                         float* __restrict__ alphaw,       // in: raw, out: w
                         const unsigned* __restrict__ amax,
                         float* __restrict__ denom) {
  int idx = blockIdx.x * blockDim.x + threadIdx.x;
  if (idx >= NE * HEADS) return;
  int e = idx >> 2, h = idx & 3;
  int d = dst[e];
  float w = expf(alphaw[idx] - ord_f32(amax[d * HEADS + h]));
  alphaw[idx] = w;
  atomicAdd(&denom[d * HEADS + h], w);
}

// ---------------- pass C: messages (ep recomputed on the fly) --------
// 256 threads = 8 waves; one wave per edge; lane handles 8 channels.
__global__ void edge_message(const int* __restrict__ src,
                             const int* __restrict__ dst,
                             const float* __restrict__ ea,     // [E,16]
                             const float* __restrict__ We,     // [16,256]
                             const float* __restrict__ xp,     // [N,256]
                             const float* __restrict__ alphaw, // [E,4] = w
                             const float* __restrict__ denom,  // [N,4]
                             float* __restrict__ aggr) {       // [N,256]
  __shared__ float ldsWe[EDIM * HC];     // 16 KB
  __shared__ float ldsEa[8][EDIM];
  int tid = threadIdx.x;
  int e0 = blockIdx.x * 8;
  for (int i = tid; i < EDIM * HC; i += 256) ldsWe[i] = We[i];
  if (tid < 8 * EDIM) {
    int j = tid >> 4, k = tid & 15;
    ldsEa[j][k] = ea[(size_t)(e0 + j) * EDIM + k];
  }
  __syncthreads();

  int wave = tid >> 5;
  int lane = tid & 31;
  int e = e0 + wave;
  int s = src[e], d = dst[e];
  int head = lane >> 3;                    // 8 channels/lane, same head
  int c0 = lane * 8;
  float coeff = alphaw[e * HEADS + head] /
                (denom[d * HEADS + head] + 1e-16f);

  float eav[EDIM];
#pragma unroll
  for (int k = 0; k < EDIM; ++k) eav[k] = ldsEa[wave][k];

  const float* xrow = xp + (size_t)s * HC + c0;
  float* arow = aggr + (size_t)d * HC + c0;
#pragma unroll
  for (int c = 0; c < 8; ++c) {
    float ep = 0.0f;
#pragma unroll
    for (int k = 0; k < EDIM; ++k) ep += eav[k] * ldsWe[k * HC + c0 + c];
    float msg = coeff * ep * xrow[c];
    atomicAdd(&arow[c], msg);
  }
}

// ---------------- fused GRU gate + LayerNorm (one wave per node) -----
__global__ void gru_ln(const float* __restrict__ gi,    // [N,192] no bias
                       const float* __restrict__ gh,    // [N,192] no bias
                       const float* __restrict__ bih,   // [192]
                       const float* __restrict__ bhh,   // [192]
                       float* __restrict__ h,           // [N,64] in/out
                       const float* __restrict__ gamma,
                       const float* __restrict__ beta,
                       float* __restrict__ out) {       // [N,64]
  int tid = threadIdx.x;
  int n = blockIdx.x * 8 + (tid >> 5);
  int lane = tid & 31;

  float val[2];
#pragma unroll
  for (int j = 0; j < 2; ++j) {
    int c = lane + 32 * j;
    size_t g = (size_t)n * GDIM + c;
    float i_r = gi[g]            + bih[c];
    float i_z = gi[g + DIM]      + bih[c + DIM];
    float i_n = gi[g + 2 * DIM]  + bih[c + 2 * DIM];
    float h_r = gh[g]            + bhh[c];
    float h_z = gh[g + DIM]      + bhh[c + DIM];
    float h_n = gh[g + 2 * DIM]  + bhh[c + 2 * DIM];
    float r = 1.0f / (1.0f + expf(-(i_r + h_r)));
    float z = 1.0f / (1.0f + expf(-(i_z + h_z)));
    float nn = tanhf(i_n + r * h_n);
    float hold = h[(size_t)n * DIM + c];
    float hv = (1.0f - z) * nn + z * hold;
    h[(size_t)n * DIM + c] = hv;
    val[j] = hv;
  }
  // layernorm over 64 channels (2 per lane, wave32 reduction)
  float s = val[0] + val[1];
#pragma unroll
  for (int off = 16; off > 0; off >>= 1) s += __shfl_xor(s, off, 32);
  float mu = s * (1.0f / DIM);
  float d0 = val[0] - mu, d1 = val[1] - mu;
  float v = d0 * d0 + d1 * d1;
#pragma unroll
  for (int off = 16; off > 0; off >>= 1) v += __shfl_xor(v, off, 32);
  float inv = rsqrtf(v * (1.0f / DIM) + 1e-5f);
#pragma unroll
  for (int j = 0; j < 2; ++j) {
    int c = lane + 32 * j;
    float dv = val[j] - mu;
    out[(size_t)n * DIM + c] = dv * inv * gamma[c] + beta[c];
  }
}

// ---------------- simple copy ----------------------------------------
__global__ void copy_f32(const float* __restrict__ a, float* __restrict__ b,
                         int count) {
  int i = blockIdx.x * blockDim.x + threadIdx.x;
  if (i < count) b[i] = a[i];
}

// =====================================================================
extern "C" void kernel_launch(void* const* d_in, const int* in_sizes, int n_in,
                              void* d_out, int out_size, void* d_ws, size_t ws_size,
                              hipStream_t stream) {
  const float* x    = (const float*)d_in[0];
  const int*   ei   = (const int*)d_in[1];     // [2,E]
  const float* ea   = (const float*)d_in[2];   // [E,16]
  const float* Wn   = (const float*)d_in[3];   // [64,256]
  const float* We   = (const float*)d_in[4];   // [16,256]
  const float* Wa   = (const float*)d_in[5];   // [1,4,192]
  const float* Ws   = (const float*)d_in[6];   // [256,64]
  const float* b    = (const float*)d_in[7];   // [64]
  const float* Wih  = (const float*)d_in[8];   // [192,64]
  const float* Whh  = (const float*)d_in[9];   // [192,64]
  const float* bih  = (const float*)d_in[10];  // [192]
  const float* bhh  = (const float*)d_in[11];  // [192]
  const float* gamma= (const float*)d_in[12];  // [64]
  const float* beta = (const float*)d_in[13];  // [64]
  float* out = (float*)d_out;

  const int* src = ei;
  const int* dst = ei + NE;

  // workspace carve-up (floats)
  float* ws = (float*)d_ws;
  float*    xp     = ws;                 ws += (size_t)NN * HC;     // 12.8M
  float*    aggr   = ws;                 ws += (size_t)NN * HC;     // 12.8M
  float*    gi     = ws;                 ws += (size_t)NN * GDIM;   // 9.6M
  float*    gh     = ws;                 ws += (size_t)NN * GDIM;   // 9.6M
  float*    alphaw = ws;                 ws += (size_t)NE * HEADS;  // 3.2M
  float*    mbuf   = ws;                 ws += (size_t)NN * DIM;    // 3.2M
  float*    hbuf   = ws;                 ws += (size_t)NN * DIM;    // 3.2M
  float*    xcbuf  = ws;                 ws += (size_t)NN * DIM;    // 3.2M
  float*    ai     = ws;                 ws += (size_t)NN * HEADS;
  float*    aj     = ws;                 ws += (size_t)NN * HEADS;
  float*    denom  = ws;                 ws += (size_t)NN * HEADS;
  unsigned* amax   = (unsigned*)ws;      ws += (size_t)NN * HEADS;
  float*    Wea    = ws;                 ws += EDIM * HEADS;

  // loop-invariant precompute
  fold_wea<<<1, 64, 0, stream>>>(We, Wa, Wea);
  copy_f32<<<(NN * DIM + 255) / 256, 256, 0, stream>>>(x, hbuf, NN * DIM);

  for (int t = 0; t < 3; ++t) {
    const float* xc_in = (t == 0) ? x : xcbuf;
    float* xc_out = (t == 2) ? out : xcbuf;

    // xp = xc @ Wn   [N,64]x[64,256]
    gemm16_f32<DIM, HC, false, false>
        <<<dim3(NN / 16, HC / 16), 32, 0, stream>>>(xc_in, Wn, nullptr, xp);

    node_alpha<<<(NN * HEADS + 255) / 256, 256, 0, stream>>>(xp, Wa, ai, aj);

    init_step<<<(NN * HC + 255) / 256, 256, 0, stream>>>(aggr, denom, amax);

    edge_alpha<<<(NE * HEADS + 255) / 256, 256, 0, stream>>>(
        src, dst, ea, Wea, ai, aj, alphaw, amax);

    edge_exp<<<(NE * HEADS + 255) / 256, 256, 0, stream>>>(
        dst, alphaw, amax, denom);

    edge_message<<<NE / 8, 256, 0, stream>>>(
        src, dst, ea, We, xp, alphaw, denom, aggr);

    // m = celu(aggr @ Ws + b)   [N,256]x[256,64]
    gemm16_f32<HC, DIM, false, true>
        <<<dim3(NN / 16, DIM / 16), 32, 0, stream>>>(aggr, Ws, b, mbuf);

    // gi = m @ Wih^T, gh = h @ Whh^T   [N,64]x[64,192]
    gemm16_f32<DIM, GDIM, true, false>
        <<<dim3(NN / 16, GDIM / 16), 32, 0, stream>>>(mbuf, Wih, nullptr, gi);
    gemm16_f32<DIM, GDIM, true, false>
        <<<dim3(NN / 16, GDIM / 16), 32, 0, stream>>>(hbuf, Whh, nullptr, gh);

    gru_ln<<<NN / 8, 256, 0, stream>>>(gi, gh, bih, bhh, hbuf, gamma, beta,
                                       xc_out);
  }
}